// Mamba_TransFormer_58772332478715
// MI455X (gfx1250) — compile-verified
//
#include <hip/hip_runtime.h>
#include <hip/hip_bf16.h>
#include <math.h>

// ---------------------------------------------------------------------------
// Mamba-TransFormer block for MI455X (gfx1250, wave32).
// GEMMs run through V_WMMA_F32_16X16X4_F32 (native fp32 matrix pipe, no
// precision loss vs fp32 reference). Aligned GEMMs use a guard-free 64x64
// tile with float4 global->LDS staging and 16 chained WMMAs per K-panel.
// SSM scan: one wave per (b,d), 2 states/lane, shfl_xor reduction.
// Workspace: ~74 MB fp32 (fits the 192 MB L2).
// ---------------------------------------------------------------------------

#define HD   256    // D
#define DIN  512
#define SS   64     // ssm_d_state
#define RR   16     // dt_rank
#define KC   4      // conv kernel
#define FFW  2048
#define BB   4
#define LL   1024
#define BL   (BB*LL)   // 4096 tokens

typedef __attribute__((ext_vector_type(2))) float v2f;
typedef __attribute__((ext_vector_type(8))) float v8f;

__device__ __forceinline__ float gelu_tanh(float x) {
  const float c0 = 0.7978845608028654f;   // sqrt(2/pi)
  float x3 = x * x * x;
  return 0.5f * x * (1.f + tanhf(c0 * (x + 0.044715f * x3)));
}

__device__ __forceinline__ float softplus_f(float x) {
  return (x > 20.f) ? x : log1pf(__expf(x));
}

__device__ __forceinline__ v8f wmma_f32(v2f a, v2f b, v8f c) {
  return __builtin_amdgcn_wmma_f32_16x16x4_f32(false, a, false, b,
                                               (short)0, c, false, false);
}

// ---------------------------------------------------------------------------
// FAST GEMM (alignment required: M%64==0, N%64==0, K%32==0).
// Block: 256 thr = 8 waves; block tile 64x64; each wave owns two 16x16 WMMA
// tiles sharing the A fragment. K panels of 32 -> 16 v_wmma per stage.
// ---------------------------------------------------------------------------
#define FTM 64
#define FTN 64
#define FTK 32
#define APAD 4   // As row stride 36 floats: float4-aligned, conflict-free frags
#define BPAD 4   // Bs row stride 68 floats: float4-aligned

__global__ __launch_bounds__(256) void kern_gemm_wmma_fast(
    const float* __restrict__ A, int lda,
    const float* __restrict__ B, int ldb,
    float* __restrict__ C, int ldc, int Kdim) {
  __shared__ float As[FTM][FTK + APAD];
  __shared__ float Bs[FTK][FTN + BPAD];
  const int tid  = threadIdx.x;
  const int wave = tid >> 5;
  const int lane = tid & 31;
  const int wm   = wave >> 1;            // 0..3
  const int nb   = (wave & 1) * 32;      // 0 or 32
  const int m0   = blockIdx.y * FTM;
  const int n0   = blockIdx.x * FTN;
  const int lm   = lane & 15;            // fragment M (A) / N (B) index
  const int kh   = (lane >> 4) << 1;     // K sub-offset 0 / 2

  v8f acc0 = {}, acc1 = {};

  for (int k0 = 0; k0 < Kdim; k0 += FTK) {
    if (k0 + FTK < Kdim) {
      __builtin_prefetch(A + (size_t)(m0 + (tid & 63)) * lda + k0 + FTK, 0, 1);
    }
    // A tile 64x32: 2 x float4 per thread, guard-free
#pragma unroll
    for (int i = 0; i < 2; ++i) {
      int idx = (tid + i * 256) * 4;
      int r = idx >> 5, c = idx & 31;
      *(float4*)&As[r][c] =
          *(const float4*)&A[(size_t)(m0 + r) * lda + k0 + c];
    }
    // B tile 32x64: 2 x float4 per thread, guard-free
#pragma unroll
    for (int i = 0; i < 2; ++i) {
      int idx = (tid + i * 256) * 4;
      int r = idx >> 6, c = idx & 63;
      *(float4*)&Bs[r][c] =
          *(const float4*)&B[(size_t)(k0 + r) * ldb + n0 + c];
    }
    __syncthreads();
#pragma unroll
    for (int kk = 0; kk < FTK; kk += 4) {
      v2f af, bf0, bf1;
      const int arow = wm * 16 + lm;
      af.x  = As[arow][kk + kh];
      af.y  = As[arow][kk + kh + 1];
      bf0.x = Bs[kk + kh][nb + lm];
      bf0.y = Bs[kk + kh + 1][nb + lm];
      bf1.x = Bs[kk + kh][nb + 16 + lm];
      bf1.y = Bs[kk + kh + 1][nb + 16 + lm];
      acc0 = wmma_f32(af, bf0, acc0);
      acc1 = wmma_f32(af, bf1, acc1);
    }
    __syncthreads();
  }

  // C/D layout: VGPR v -> M = (lane/16)*8 + v, N = lane%16
  const int mrow = m0 + wm * 16 + (lane >> 4) * 8;
  const int nc   = n0 + nb + lm;
#pragma unroll
  for (int v = 0; v < 8; ++v) {
    C[(size_t)(mrow + v) * ldc + nc]      = acc0[v];
    C[(size_t)(mrow + v) * ldc + nc + 16] = acc1[v];
  }
}

// ---------------------------------------------------------------------------
// GUARDED GEMM (any M/N/K; used for N=144 and K=16 cases).
// Block tile 64x32, one 16x16 WMMA tile per wave, K panels of 16.
// ---------------------------------------------------------------------------
#define TM 64
#define TN 32
#define TK 16

__global__ __launch_bounds__(256) void kern_gemm_wmma_guard(
    const float* __restrict__ A, int lda,
    const float* __restrict__ B, int ldb,
    float* __restrict__ C, int ldc,
    int M, int N, int Kdim) {
  __shared__ float As[TM][TK + 1];
  __shared__ float Bs[TK][TN + 1];
  const int tid  = threadIdx.x;
  const int wave = tid >> 5;
  const int lane = tid & 31;
  const int wm   = wave >> 1;
  const int wn   = wave & 1;
  const int m0   = blockIdx.y * TM;
  const int n0   = blockIdx.x * TN;
  const int lm   = lane & 15;
  const int kh   = (lane >> 4) << 1;

  v8f acc = {};

  for (int k0 = 0; k0 < Kdim; k0 += TK) {
#pragma unroll
    for (int i = 0; i < 4; ++i) {
      int idx = tid + i * 256;
      int r = idx >> 4, c = idx & 15;
      int gm = m0 + r, gk = k0 + c;
      As[r][c] = (gm < M && gk < Kdim) ? A[(size_t)gm * lda + gk] : 0.f;
    }
#pragma unroll
    for (int i = 0; i < 2; ++i) {
      int idx = tid + i * 256;
      int r = idx >> 5, c = idx & 31;
      int gk = k0 + r, gn = n0 + c;
      Bs[r][c] = (gk < Kdim && gn < N) ? B[(size_t)gk * ldb + gn] : 0.f;
    }
    __syncthreads();
#pragma unroll
    for (int kk = 0; kk < TK; kk += 4) {
      v2f af, bf;
      af.x = As[wm * 16 + lm][kk + kh];
      af.y = As[wm * 16 + lm][kk + kh + 1];
      bf.x = Bs[kk + kh][wn * 16 + lm];
      bf.y = Bs[kk + kh + 1][wn * 16 + lm];
      acc = wmma_f32(af, bf, acc);
    }
    __syncthreads();
  }

  const int mrow = m0 + wm * 16 + (lane >> 4) * 8;
  const int ncol = n0 + wn * 16 + lm;
  if (ncol < N) {
#pragma unroll
    for (int v = 0; v < 8; ++v) {
      int gm = mrow + v;
      if (gm < M) C[(size_t)gm * ldc + ncol] = acc[v];
    }
  }
}

// ---------------------------------------------------------------------------
// LayerNorm over last dim (=256). One block (256 thr) per row.
// ---------------------------------------------------------------------------
__global__ __launch_bounds__(256) void kern_ln(
    const float* __restrict__ x, const float* __restrict__ g,
    const float* __restrict__ b, float* __restrict__ o) {
  __shared__ float red_s[8];
  __shared__ float red_q[8];
  const int row = blockIdx.x;
  const int tid = threadIdx.x;
  const int wave = tid >> 5, lane = tid & 31;
  float v = x[(size_t)row * HD + tid];
  float s = v, q = v * v;
#pragma unroll
  for (int off = 16; off; off >>= 1) {
    s += __shfl_xor(s, off, 32);
    q += __shfl_xor(q, off, 32);
  }
  if (lane == 0) { red_s[wave] = s; red_q[wave] = q; }
  __syncthreads();
  float sum = 0.f, sq = 0.f;
#pragma unroll
  for (int i = 0; i < 8; ++i) { sum += red_s[i]; sq += red_q[i]; }
  float mean = sum * (1.f / HD);
  float var  = sq * (1.f / HD) - mean * mean;
  float inv  = rsqrtf(var + 1e-5f);
  o[(size_t)row * HD + tid] = (v - mean) * inv * g[tid] + b[tid];
}

// ---------------------------------------------------------------------------
// Causal depthwise conv (K=4) + bias + tanh-GELU.  u0,u: [B,L,DIN]
// ---------------------------------------------------------------------------
__global__ __launch_bounds__(256) void kern_conv_gelu(
    const float* __restrict__ u0, const float* __restrict__ w,
    const float* __restrict__ bias, float* __restrict__ u, int total) {
  int i = blockIdx.x * blockDim.x + threadIdx.x;
  if (i >= total) return;
  int d  = i % DIN;
  int bl = i / DIN;
  int l  = bl % LL;
  float s = bias[d];
#pragma unroll
  for (int k = 0; k < KC; ++k) {
    int ls = l - (KC - 1) + k;
    if (ls >= 0) s += u0[(size_t)(bl - (KC - 1) + k) * DIN + d] * w[d * KC + k];
  }
  u[i] = gelu_tanh(s);
}

__global__ __launch_bounds__(256) void kern_softplus_bias(
    const float* __restrict__ x, const float* __restrict__ bdt,
    float* __restrict__ o, int total) {
  int i = blockIdx.x * blockDim.x + threadIdx.x;
  if (i >= total) return;
  o[i] = softplus_f(x[i] + bdt[i % DIN]);
}

__global__ __launch_bounds__(256) void kern_bias_gelu(
    float* __restrict__ x, const float* __restrict__ bias, int cols, int total) {
  int i = blockIdx.x * blockDim.x + threadIdx.x;
  if (i >= total) return;
  x[i] = gelu_tanh(x[i] + bias[i % cols]);
}

// out = a*x + b*y  (out may alias x or y)
__global__ __launch_bounds__(256) void kern_axpby(
    float* __restrict__ out, const float* __restrict__ x,
    const float* __restrict__ y, float a, float b, int n) {
  int i = blockIdx.x * blockDim.x + threadIdx.x;
  if (i < n) out[i] = a * x[i] + b * y[i];
}

// out = x + bias[col] + res
__global__ __launch_bounds__(256) void kern_bias_add2(
    float* __restrict__ out, const float* __restrict__ x,
    const float* __restrict__ bias, const float* __restrict__ res,
    int cols, int total) {
  int i = blockIdx.x * blockDim.x + threadIdx.x;
  if (i >= total) return;
  out[i] = x[i] + bias[i % cols] + res[i];
}

// ---------------------------------------------------------------------------
// Selective-scan. One wave per (b,d); 64 states -> 2/lane; shfl_xor reduce.
// u,dt,y: [B,L,DIN]; dbc: [B,L,144] (B=cols 16:80, C=cols 80:144);
// A_log:[DIN,S]; Dp:[DIN].
// ---------------------------------------------------------------------------
__global__ __launch_bounds__(256) void kern_scan(
    const float* __restrict__ u, const float* __restrict__ dt,
    const float* __restrict__ dbc, const float* __restrict__ A_log,
    const float* __restrict__ Dp, float* __restrict__ y) {
  const int wave = threadIdx.x >> 5;
  const int lane = threadIdx.x & 31;
  const int bd = blockIdx.x * 8 + wave;   // 0..B*DIN-1
  const int b = bd >> 9;                  // /DIN
  const int d = bd & (DIN - 1);
  const int s0 = lane, s1 = lane + 32;
  const float A0 = -__expf(A_log[d * SS + s0]);
  const float A1 = -__expf(A_log[d * SS + s1]);
  const float dp = Dp[d];
  float h0 = 0.f, h1 = 0.f;
  const float* ud  = u  + (size_t)b * LL * DIN + d;
  const float* dtd = dt + (size_t)b * LL * DIN + d;
  const float* row = dbc + (size_t)b * LL * (RR + 2 * SS);
  float* yd = y + (size_t)b * LL * DIN + d;
  for (int t = 0; t < LL; ++t) {
    float dtv = dtd[(size_t)t * DIN];
    float uv  = ud[(size_t)t * DIN];
    const float* r = row + t * (RR + 2 * SS);
    float B0 = r[RR + s0],      B1 = r[RR + s1];
    float C0 = r[RR + SS + s0], C1 = r[RR + SS + s1];
    float du = dtv * uv;
    h0 = __expf(dtv * A0) * h0 + du * B0;
    h1 = __expf(dtv * A1) * h1 + du * B1;
    float yp = h0 * C0 + h1 * C1;
#pragma unroll
    for (int off = 16; off; off >>= 1) yp += __shfl_xor(yp, off, 32);
    if (lane == 0) yd[(size_t)t * DIN] = yp + uv * dp;
  }
}

// ---------------------------------------------------------------------------
// Host: orchestrate the whole block on `stream`.
// d_in order (depth-first dict order):
//  0 content, 1 style,
//  per-VSS (c then s): ln_g,ln_b,Win,conv_w,conv_b,Wx,Wdt,bdt,A_log,Dp,Wout
//  then W1,b1,W2,b2, ln1_g,ln1_b, ln2_g,ln2_b, ln3_g,ln3_b
// ---------------------------------------------------------------------------
extern "C" void kernel_launch(void* const* d_in, const int* in_sizes, int n_in,
                              void* d_out, int out_size, void* d_ws, size_t ws_size,
                              hipStream_t stream) {
  (void)in_sizes; (void)n_in; (void)out_size; (void)ws_size;
  const float* content = (const float*)d_in[0];
  const float* style   = (const float*)d_in[1];

  // workspace layout (floats); ~74 MB total
  float* ws  = (float*)d_ws;
  float* xn  = ws;                        // 4096*256
  float* u0  = xn  + (size_t)BL * HD;     // 4096*512 (also reused for dt)
  float* uu  = u0  + (size_t)BL * DIN;    // 4096*512
  float* dbc = uu  + (size_t)BL * DIN;    // 4096*144
  float* yy  = dbc + (size_t)BL * (RR + 2 * SS);  // 4096*512
  float* t0  = yy  + (size_t)BL * DIN;    // 4096*256
  float* tgt = t0  + (size_t)BL * HD;     // 4096*256
  float* ff  = tgt + (size_t)BL * HD;     // 4096*2048

  // aligned fast GEMM: M%64==0, N%64==0, K%32==0
  auto gemm_fast = [&](const float* A, int lda, const float* B, int ldb,
                       float* C, int ldc, int M, int N, int Kdim) {
    dim3 g(N / FTN, M / FTM);
    kern_gemm_wmma_fast<<<g, 256, 0, stream>>>(A, lda, B, ldb, C, ldc, Kdim);
  };
  auto gemm_guard = [&](const float* A, int lda, const float* B, int ldb,
                        float* C, int ldc, int M, int N, int Kdim) {
    dim3 g((N + TN - 1) / TN, M / TM);
    kern_gemm_wmma_guard<<<g, 256, 0, stream>>>(A, lda, B, ldb, C, ldc, M, N, Kdim);
  };

  auto vss = [&](const float* x, int pb, float* out_gemm) {
    const float* ln_g  = (const float*)d_in[pb + 0];
    const float* ln_b  = (const float*)d_in[pb + 1];
    const float* Win   = (const float*)d_in[pb + 2];
    const float* convw = (const float*)d_in[pb + 3];
    const float* convb = (const float*)d_in[pb + 4];
    const float* Wx    = (const float*)d_in[pb + 5];
    const float* Wdt   = (const float*)d_in[pb + 6];
    const float* bdt   = (const float*)d_in[pb + 7];
    const float* A_log = (const float*)d_in[pb + 8];
    const float* Dp    = (const float*)d_in[pb + 9];
    const float* Wout  = (const float*)d_in[pb + 10];

    kern_ln<<<BL, 256, 0, stream>>>(x, ln_g, ln_b, xn);
    gemm_fast(xn, HD, Win, DIN, u0, DIN, BL, DIN, HD);            // xn @ Win
    {
      int total = BL * DIN;
      kern_conv_gelu<<<(total + 255) / 256, 256, 0, stream>>>(u0, convw, convb, uu, total);
    }
    gemm_guard(uu, DIN, Wx, RR + 2 * SS, dbc, RR + 2 * SS,
               BL, RR + 2 * SS, DIN);                             // u @ Wx (N=144)
    gemm_guard(dbc, RR + 2 * SS, Wdt, DIN, u0, DIN,
               BL, DIN, RR);                                      // dbc[:, :16] @ Wdt (K=16)
    {
      int total = BL * DIN;
      kern_softplus_bias<<<(total + 255) / 256, 256, 0, stream>>>(u0, bdt, u0, total);
    }
    kern_scan<<<(BB * DIN) / 8, 256, 0, stream>>>(uu, u0, dbc, A_log, Dp, yy);
    gemm_fast(yy, DIN, Wout, HD, out_gemm, HD, BL, HD, DIN);      // y @ Wout
  };

  const int NTOK = BL * HD;

  // --- content branch: tgt = LN(2*content + yc@Wout, ln1) ---
  vss(content, 2, t0);
  kern_axpby<<<(NTOK + 255) / 256, 256, 0, stream>>>(xn, content, t0, 2.f, 1.f, NTOK);
  kern_ln<<<BL, 256, 0, stream>>>(xn, (const float*)d_in[28], (const float*)d_in[29], tgt);

  // --- style branch: tgt += LN(2*style + ys@Wout, ln2) ---
  vss(style, 13, t0);
  kern_axpby<<<(NTOK + 255) / 256, 256, 0, stream>>>(xn, style, t0, 2.f, 1.f, NTOK);
  kern_ln<<<BL, 256, 0, stream>>>(xn, (const float*)d_in[30], (const float*)d_in[31], t0);
  kern_axpby<<<(NTOK + 255) / 256, 256, 0, stream>>>(tgt, tgt, t0, 1.f, 1.f, NTOK);

  // --- FFN: LN(tgt + gelu(tgt@W1+b1)@W2 + b2, ln3) ---
  gemm_fast(tgt, HD, (const float*)d_in[24], FFW, ff, FFW, BL, FFW, HD);
  {
    int total = BL * FFW;
    kern_bias_gelu<<<(total + 255) / 256, 256, 0, stream>>>(ff, (const float*)d_in[25], FFW, total);
  }
  gemm_fast(ff, FFW, (const float*)d_in[26], HD, t0, HD, BL, HD, FFW);
  kern_bias_add2<<<(NTOK + 255) / 256, 256, 0, stream>>>(xn, t0, (const float*)d_in[27], tgt, HD, NTOK);
  kern_ln<<<BL, 256, 0, stream>>>(xn, (const float*)d_in[32], (const float*)d_in[33], (float*)d_out);
}